// SuctionNet_11252814315966
// MI455X (gfx1250) — compile-verified
//
#include <hip/hip_runtime.h>

typedef float v2f __attribute__((ext_vector_type(2)));
typedef float v8f __attribute__((ext_vector_type(8)));

#define FEAT_DIM 256

// One wave32 per 16 points.
//   A (16x4 f32)  = w_scoring slice; only matrix rows M=0,1 are meaningful,
//                   rows 2..15 hold replicated garbage that is never read out.
//   B (4x16 f32)  = gathered feature columns for 16 points.
//   C (16x16 f32) accumulates over K=256 in 64 steps of V_WMMA_F32_16X16X4_F32.
// C VGPR0/lanes0-15 = seal dot for the 16 points, VGPR1/lanes0-15 = wrench dot.
__global__ void __launch_bounds__(256)
suction_score_wmma(const float* __restrict__ feats,
                   const int*   __restrict__ idx,
                   const float* __restrict__ w,
                   const float* __restrict__ bias,
                   float*       __restrict__ out,
                   int npts) {
  const int wave   = (int)((blockIdx.x * blockDim.x + threadIdx.x) >> 5);
  const int lane   = (int)(threadIdx.x & 31u);
  const int ntiles = npts >> 4;                 // npts is a multiple of 16
  if (wave >= ntiles) return;                   // wave-uniform exit: EXEC stays all-1s

  const int base = wave << 4;
  const int n    = lane & 15;                   // point-in-tile / matrix column
  const int hi   = lane >> 4;                   // selects K pair {0,1} vs {2,3}

  int p = base + n;
  if (p >= npts) p = npts - 1;                  // per-lane address clamp only
  const long long row = (long long)idx[p];

  const float* brow = feats + row * FEAT_DIM + 2 * hi;       // feature stream
  const float* arow = w + (long long)(n & 1) * FEAT_DIM + 2 * hi; // w rows 0/1

  __builtin_prefetch(brow, 0, 1);               // global_prefetch_b8 on the row

  v8f c = {};
#pragma unroll 8
  for (int k = 0; k < FEAT_DIM; k += 4) {
    v2f a = *(const v2f*)(arow + k);
    v2f b = *(const v2f*)(brow + k);
    // 8 args: (neg_a, A, neg_b, B, c_mod, C, reuse_a, reuse_b)
    c = __builtin_amdgcn_wmma_f32_16x16x4_f32(
        /*neg_a=*/false, a, /*neg_b=*/false, b,
        /*c_mod=*/(short)0, c, /*reuse_a=*/false, /*reuse_b=*/false);
  }

  // Lanes 0..15 hold N=0..15; c[0] is M=0 (seal), c[1] is M=1 (wrench).
  if (lane < 16) {
    const int pt = base + lane;
    if (pt < npts) {
      out[pt]        = c[0] + bias[0];
      out[npts + pt] = c[1] + bias[1];
    }
  }
}

extern "C" void kernel_launch(void* const* d_in, const int* in_sizes, int n_in,
                              void* d_out, int out_size, void* d_ws, size_t ws_size,
                              hipStream_t stream) {
  const float* feats = (const float*)d_in[0];   // (150000, 256) f32
  const int*   idx   = (const int*)d_in[1];     // (200000,) indices
  const float* w     = (const float*)d_in[2];   // (2, 256) f32
  const float* bias  = (const float*)d_in[3];   // (2,) f32
  float*       out   = (float*)d_out;           // [seal(200000) | wrench(200000)]

  const int npts   = in_sizes[1];               // 200000
  const int ntiles = (npts + 15) / 16;          // one wave32 per tile
  const int threads = 256;                      // 8 waves per block
  const int blocks  = (ntiles * 32 + threads - 1) / threads;

  suction_score_wmma<<<blocks, threads, 0, stream>>>(feats, idx, w, bias, out, npts);
}